// SelfAttentionLayer2_41764261987262
// MI455X (gfx1250) — compile-verified
//
#include <hip/hip_runtime.h>
#include <hip/hip_bf16.h>
#include <math.h>

// ---------------------------------------------------------------------------
// Problem constants (from reference):
//   x      : [B=64, T=8192, D=80] fp32      (D = H*HD = 8*10)
//   out    : [B, H, T] fp32  (gate == 1 exactly: softmax over a size-1 axis)
// Memory-bound: ~336 MB traffic -> ~14us at 23.3 TB/s. x (160MB) fits in the
// 192MB L2, so pass 2 re-reads mostly hit L2.
//
// Pass 1 uses V_WMMA_F32_16X16X4_F32 as a reduction engine:
//   A = ones(16x4)  =>  D[m,n] += sum_k B[k,n]  (column sums, 4 T-rows/step)
// ---------------------------------------------------------------------------

typedef __attribute__((ext_vector_type(2))) float v2f;
typedef __attribute__((ext_vector_type(8))) float v8f;

#define B_   64
#define T_   8192
#define H_   8
#define HD_  10
#define D_   80               // model dim
#define NT_  32               // T-chunks for pass 1 partials
#define ROWS_ (T_ / NT_)      // 256 rows per chunk

// ---------------------------------------------------------------------------
// Pass 1: per-(b, t-chunk) column partial sums via WMMA f32 16x16x4.
// Block = 160 threads = 5 waves; wave w owns 16 columns [w*16, w*16+16).
// Each wave step consumes 4 consecutive T-rows:
//   B-matrix VGPR0: lanes 0-15 -> row t+0, lanes 16-31 -> row t+2
//   B-matrix VGPR1: lanes 0-15 -> row t+1, lanes 16-31 -> row t+3
//   column n = lane & 15.  With A = ones, D[0,n] = running column-n sum.
// ---------------------------------------------------------------------------
__global__ void __launch_bounds__(160)
colsum_wmma_kernel(const float* __restrict__ x, float* __restrict__ partial) {
    const int b    = blockIdx.x;
    const int tc   = blockIdx.y;
    const int wave = threadIdx.x >> 5;    // 0..4  -> column group
    const int lane = threadIdx.x & 31;
    const int half = lane >> 4;           // 0 or 1
    const int n    = lane & 15;           // column within group
    const int c0   = wave * 16;
    const int t0   = tc * ROWS_;

    // lane reads rows (t0 + 2*half) and (t0 + 2*half + 1), column c0+n
    const float* p = x + ((size_t)b * T_ + (size_t)(t0 + half * 2)) * D_ + c0 + n;

    v2f a; a.x = 1.0f; a.y = 1.0f;        // ones A-matrix (16x4)
    v8f acc = {};                          // 16x16 f32 accumulator

    #pragma unroll 4
    for (int tt = 0; tt < ROWS_; tt += 4) {
        v2f bv;
        bv.x = p[0];                       // row t0+tt+2*half
        bv.y = p[D_];                      // row t0+tt+2*half+1
        // 8 args: (neg_a, A, neg_b, B, c_mod, C, reuse_a, reuse_b)
        acc = __builtin_amdgcn_wmma_f32_16x16x4_f32(
            false, a, false, bv, (short)0, acc, false, false);
        p += 4 * D_;
    }

    // Every m-row of D holds the same column sums; lanes 16-31 duplicate 0-15.
    if (lane < 16) {
        partial[((size_t)b * NT_ + tc) * D_ + c0 + n] = acc[0];
    }
}

// ---------------------------------------------------------------------------
// Pass 1b: reduce NT_ partials -> ksum[b][c]   (5120 outputs, tiny kernel)
// ---------------------------------------------------------------------------
__global__ void reduce_partials_kernel(const float* __restrict__ partial,
                                       float* __restrict__ ksum) {
    const int i = blockIdx.x * blockDim.x + threadIdx.x;
    if (i >= B_ * D_) return;
    const int b = i / D_;
    const int c = i % D_;
    float s = 0.0f;
    #pragma unroll
    for (int tc = 0; tc < NT_; ++tc)
        s += partial[((size_t)b * NT_ + tc) * D_ + c];
    ksum[i] = s;
}

// ---------------------------------------------------------------------------
// Pass 2: fused tanh -> softmax(size 10) -> weighted sum.
// Block = 256 threads handles one b and 32 t's (all 8 heads).
// Read mapping  : tid -> (h = tid&7, t = t0 + tid>>3)  => 8 consecutive lanes
//                 cover one contiguous 320B row segment (coalesced reads).
// Write mapping : via LDS tile -> consecutive lanes write consecutive t
//                 (128B per wave, coalesced writes).
// ---------------------------------------------------------------------------
__global__ void __launch_bounds__(256)
attn_out_kernel(const float* __restrict__ x, const float* __restrict__ ksum,
                float* __restrict__ out) {
    __shared__ float s_ks[D_];
    __shared__ float tile[H_ * 32];

    const int b   = blockIdx.x;
    const int t0  = blockIdx.y * 32;
    const int tid = threadIdx.x;

    const int h  = tid & 7;
    const int ti = tid >> 3;
    const int t  = t0 + ti;
    const float* q = x + ((size_t)b * T_ + t) * D_ + h * HD_;

    __builtin_prefetch(q, 0, 0);          // global_prefetch_b8 (overlap LDS fill)

    if (tid < D_) s_ks[tid] = ksum[b * D_ + tid];
    __syncthreads();

    // Load 10 floats as 5 x float2 (40B offsets -> 8B aligned).
    float qv[HD_];
    #pragma unroll
    for (int d = 0; d < HD_; d += 2) {
        const float2 v = *reinterpret_cast<const float2*>(q + d);
        qv[d] = v.x; qv[d + 1] = v.y;
    }

    const float inv_sqrt_hd = 0.31622776601683794f;   // 1/sqrt(10)
    float esum = 0.0f, num = 0.0f;
    #pragma unroll
    for (int d = 0; d < HD_; ++d) {
        const float z  = tanhf(qv[d] * s_ks[h * HD_ + d] * inv_sqrt_hd);
        const float ex = expf(z);          // z in [-1,1]: no overflow concern
        esum += ex;
        num  += ex * qv[d];
    }
    tile[h * 32 + ti] = num / esum;        // = sum_d softmax_d * q_d
    __syncthreads();

    const int h2 = tid >> 5;
    const int t2 = tid & 31;
    out[((size_t)b * H_ + h2) * T_ + (size_t)(t0 + t2)] = tile[h2 * 32 + t2];
}

// ---------------------------------------------------------------------------
// Launch: 3 deterministic kernels (no float atomics), all on `stream`.
// ws layout: [0, 163840) partials  | [163840, 168960) ksum   (floats, ~660KB)
// d_in[1] (w_proj) and d_in[2] (b_proj) are mathematically unused:
// softmax over the size-1 gate axis is identically 1.
// ---------------------------------------------------------------------------
extern "C" void kernel_launch(void* const* d_in, const int* in_sizes, int n_in,
                              void* d_out, int out_size, void* d_ws, size_t ws_size,
                              hipStream_t stream) {
    const float* x  = (const float*)d_in[0];
    float* ws       = (float*)d_ws;
    float* partial  = ws;                                   // B*NT*D floats
    float* ksum     = ws + (size_t)B_ * NT_ * D_;           // B*D floats
    float* out      = (float*)d_out;

    colsum_wmma_kernel<<<dim3(B_, NT_), 160, 0, stream>>>(x, partial);
    reduce_partials_kernel<<<(B_ * D_ + 255) / 256, 256, 0, stream>>>(partial, ksum);
    attn_out_kernel<<<dim3(B_, T_ / 32), 256, 0, stream>>>(x, ksum, out);
}